// Tomato_15582141350410
// MI455X (gfx1250) — compile-verified
//
#include <hip/hip_runtime.h>

// CDNA5 / gfx1250 — wave32, WMMA f32 16x16x4.
// Pipeline: kNN-KDE density -> stable argsort by density -> k=15 NN graph.
//
// Workspace layout (floats): [sq N][dens N][sqs N][xs N*64][maxbits 1u]
// Output layout: [dens_sorted N floats][rips_idxs N*15 int32 (bit-stored)]

typedef __attribute__((ext_vector_type(2))) float v2f;
typedef __attribute__((ext_vector_type(8))) float v8f;

#define K_RIPS 15
#define K_PAD 16  // list length incl. +inf sentinel for the 4-way merge
#define DENS_SCALE (1.0f / (100.0f * 0.5f))  // 1/(K_KDE*SCALE)

// ---------------------------------------------------------------------------
// 16x16 fp32 Gram tile via V_WMMA_F32_16X16X4_F32.
// A lane layout (16x4 f32): lane L -> row (L&15), k-base 2*(L>>4).
// B lane layout (4x16 f32): lane L -> col (L&15), k-base 2*(L>>4).
// C/D layout: lane L, elem v -> G[v + 8*(L>>4)][L&15].
// K=64 is split into two independent 8-deep WMMA chains so the scheduler can
// interleave them while B-operand loads are in flight; summed at the end.
// ---------------------------------------------------------------------------
__device__ inline v8f gram_tile16(const float* __restrict__ X, int r0, int c0, int lane) {
    const int m = lane & 15;
    const int h = lane >> 4;
    const float* ap = X + (size_t)(r0 + m) * 64 + 2 * h;
    const float* bp = X + (size_t)(c0 + m) * 64 + 2 * h;
    v8f cA = {0.f, 0.f, 0.f, 0.f, 0.f, 0.f, 0.f, 0.f};
    v8f cB = {0.f, 0.f, 0.f, 0.f, 0.f, 0.f, 0.f, 0.f};
#pragma unroll
    for (int k = 0; k < 32; k += 4) {
        v2f a0 = *(const v2f*)(ap + k);
        v2f b0 = *(const v2f*)(bp + k);
        cA = __builtin_amdgcn_wmma_f32_16x16x4_f32(false, a0, false, b0, (short)0, cA,
                                                   false, false);
        v2f a1 = *(const v2f*)(ap + 32 + k);
        v2f b1 = *(const v2f*)(bp + 32 + k);
        cB = __builtin_amdgcn_wmma_f32_16x16x4_f32(false, a1, false, b1, (short)0, cB,
                                                   false, false);
    }
    return cA + cB;
}

// ---------------------------------------------------------------------------
// Per-point squared norms.
// ---------------------------------------------------------------------------
__global__ void __launch_bounds__(256) norms_kernel(const float* __restrict__ x,
                                                    float* __restrict__ sq, int N) {
    int i = blockIdx.x * blockDim.x + threadIdx.x;
    if (i >= N) return;
    const float* row = x + (size_t)i * 64;
    float s = 0.f;
#pragma unroll
    for (int k = 0; k < 64; k += 4) {
        float4 v = *(const float4*)(row + k);
        s += v.x * v.x + v.y * v.y + v.z * v.z + v.w * v.w;
    }
    sq[i] = s;
}

// ---------------------------------------------------------------------------
// Fused WMMA distance + KDE accumulation. One block = 16 query rows,
// 4 waves stripe the 1000 column tiles. exp tail beyond top-100 is < fp32
// epsilon relative to the self term, so summing all columns == reference.
// ---------------------------------------------------------------------------
__global__ void __launch_bounds__(128) kde_kernel(const float* __restrict__ x,
                                                  const float* __restrict__ sq,
                                                  float* __restrict__ dens, int N) {
    __shared__ float sdens[16];
    const int tid  = threadIdx.x;
    const int wave = tid >> 5;
    const int lane = tid & 31;
    const int h    = lane >> 4;
    const int n    = lane & 15;
    const int r0   = blockIdx.x << 4;

    if (tid < 16) sdens[tid] = 0.f;
    __syncthreads();

    float sqr[8];
#pragma unroll
    for (int v = 0; v < 8; ++v) sqr[v] = sq[r0 + v + 8 * h];

    float acc[8] = {0.f, 0.f, 0.f, 0.f, 0.f, 0.f, 0.f, 0.f};
    const int NT = N >> 4;
    for (int ct = wave; ct < NT; ct += 4) {
        const int c0 = ct << 4;
        __builtin_prefetch(x + (size_t)(c0 + 64 + n) * 64, 0, 1);  // next batch, L2
        v8f g = gram_tile16(x, r0, c0, lane);
        float sqc = sq[c0 + n];
#pragma unroll
        for (int v = 0; v < 8; ++v) {
            float d2 = fmaxf(sqr[v] + sqc - 2.0f * g[v], 0.0f);
            acc[v] += __expf(-2.0f * d2);  // exp(-d2 / 0.5)
        }
    }
#pragma unroll
    for (int v = 0; v < 8; ++v) atomicAdd(&sdens[v + 8 * h], acc[v]);  // ds_add_f32
    __syncthreads();
    if (tid < 16) dens[r0 + tid] = sdens[tid] * DENS_SCALE;
}

// ---------------------------------------------------------------------------
// Global max of (positive) densities via uint atomicMax on float bits.
// ---------------------------------------------------------------------------
__global__ void __launch_bounds__(256) max_kernel(const float* __restrict__ dens,
                                                  unsigned* __restrict__ maxbits, int N) {
    __shared__ float red[256];
    const int tid = threadIdx.x;
    float m = 0.f;
    for (int i = blockIdx.x * 256 + tid; i < N; i += gridDim.x * 256)
        m = fmaxf(m, dens[i]);
    red[tid] = m;
    __syncthreads();
    for (int s = 128; s > 0; s >>= 1) {
        if (tid < s) red[tid] = fmaxf(red[tid], red[tid + s]);
        __syncthreads();
    }
    if (tid == 0) atomicMax(maxbits, __float_as_uint(red[0]));
}

// ---------------------------------------------------------------------------
// Stable argsort by rank-counting. The entire 64 KB density array is cached
// in LDS (320 KB/WGP); the scan reads it as float4 (ds_load_b128 broadcast).
// rank(i) = #{j : d[j]<d[i] or (d[j]==d[i] and j<i)}.
// Also permutes points + norms and writes normalized sorted densities.
// ---------------------------------------------------------------------------
__global__ void __launch_bounds__(256) rank_kernel(const float* __restrict__ x,
                                                   const float* __restrict__ sq,
                                                   const float* __restrict__ dens,
                                                   const unsigned* __restrict__ maxbits,
                                                   float* __restrict__ xs,
                                                   float* __restrict__ sqs,
                                                   float* __restrict__ dens_sorted, int N) {
    extern __shared__ float sd[];
    const int tid = threadIdx.x;
    for (int j = tid; j < N; j += blockDim.x) sd[j] = dens[j];
    __syncthreads();

    const int i = blockIdx.x * blockDim.x + tid;
    if (i >= N) return;
    const float di = sd[i];
    const float4* sd4 = (const float4*)sd;
    int rank = 0;
    for (int j4 = 0; j4 < (N >> 2); ++j4) {   // N % 4 == 0
        float4 dj = sd4[j4];
        const int j = j4 << 2;
        rank += (int)(dj.x < di) | ((int)(dj.x == di) & (int)(j + 0 < i));
        rank += (int)(dj.y < di) | ((int)(dj.y == di) & (int)(j + 1 < i));
        rank += (int)(dj.z < di) | ((int)(dj.z == di) & (int)(j + 2 < i));
        rank += (int)(dj.w < di) | ((int)(dj.w == di) & (int)(j + 3 < i));
    }
    const float inv = 1.0f / __uint_as_float(*maxbits);
    dens_sorted[rank] = di * inv;
    sqs[rank] = sq[i];
    const float* srcr = x + (size_t)i * 64;
    float* dstr = xs + (size_t)rank * 64;
#pragma unroll
    for (int k = 0; k < 64; k += 4) *(float4*)(dstr + k) = *(const float4*)(srcr + k);
}

// ---------------------------------------------------------------------------
// Fused WMMA distance + top-15 selection on density-sorted points.
// 4 waves fill a 16x64 distance slab in LDS. Selection: top_k's tie-break
// (lower index first among equal values) is exactly lexicographic (d, idx)
// order, so the scan can be parallelized: 64 worker threads each own
// (row, column-segment) and keep a sorted (d, idx) list; a final exact 4-way
// lexicographic merge per row recovers the global top-15. Lists are padded
// with a +inf sentinel entry so merge head pointers never run off the end.
// ---------------------------------------------------------------------------
__global__ void __launch_bounds__(128) rips_kernel(const float* __restrict__ xs,
                                                   const float* __restrict__ sqs,
                                                   int* __restrict__ out, int N) {
    __shared__ float tile[16][65];          // +1 pad: conflict-free row reads
    __shared__ float bestd[4][16][K_PAD];
    __shared__ int   besti[4][16][K_PAD];
    const int tid  = threadIdx.x;
    const int wave = tid >> 5;
    const int lane = tid & 31;
    const int h    = lane >> 4;
    const int n    = lane & 15;
    const int r0   = blockIdx.x << 4;

    if (tid < 64) {
        const int seg = tid >> 4, row = tid & 15;
#pragma unroll
        for (int j = 0; j < K_PAD; ++j) {
            bestd[seg][row][j] = __builtin_inff();
            besti[seg][row][j] = 0x7fffffff;
        }
    }
    float sqr[8];
#pragma unroll
    for (int v = 0; v < 8; ++v) sqr[v] = sqs[r0 + v + 8 * h];
    __syncthreads();

    const int NT = N >> 4;
    for (int base = 0; base < NT; base += 4) {
        const int ct = base + wave;             // wave-uniform predicate: EXEC stays full
        if (ct < NT) {
            const int c0 = ct << 4;
            v8f g = gram_tile16(xs, r0, c0, lane);
            float sqc = sqs[c0 + n];
#pragma unroll
            for (int v = 0; v < 8; ++v)
                tile[v + 8 * h][(wave << 4) + n] =
                    fmaxf(sqr[v] + sqc - 2.0f * g[v], 0.0f);
        } else {
#pragma unroll
            for (int v = 0; v < 8; ++v)
                tile[v + 8 * h][(wave << 4) + n] = __builtin_inff();
        }
        __syncthreads();

        if (tid < 64) {
            const int seg = tid >> 4, row = tid & 15;
            float* bd = bestd[seg][row];
            int*   bi = besti[seg][row];
            const int cbase = (base + seg) << 4;
            for (int j = 0; j < 16; ++j) {
                const float d = tile[row][(seg << 4) + j];
                const int   c = cbase + j;
                const float th = bd[K_RIPS - 1];
                if (d < th || (d == th && c < bi[K_RIPS - 1])) {
                    int p = K_RIPS - 1;
                    while (p > 0 &&
                           (bd[p - 1] > d || (bd[p - 1] == d && bi[p - 1] > c))) {
                        bd[p] = bd[p - 1];
                        bi[p] = bi[p - 1];
                        --p;
                    }
                    bd[p] = d;
                    bi[p] = c;
                }
            }
        }
        __syncthreads();
    }

    if (tid < 16) {                      // exact 4-way lexicographic (d,idx) merge
        const int row = tid;
        int p0 = 0, p1 = 0, p2 = 0, p3 = 0;
        int* orow = out + (size_t)(r0 + row) * K_RIPS;
#pragma unroll
        for (int j = 0; j < K_RIPS; ++j) {
            float d0 = bestd[0][row][p0], d1 = bestd[1][row][p1];
            float d2 = bestd[2][row][p2], d3 = bestd[3][row][p3];
            int i0 = besti[0][row][p0], i1 = besti[1][row][p1];
            int i2 = besti[2][row][p2], i3 = besti[3][row][p3];
            float da; int ia, sa;
            if (d1 < d0 || (d1 == d0 && i1 < i0)) { da = d1; ia = i1; sa = 1; }
            else                                  { da = d0; ia = i0; sa = 0; }
            float db; int ib, sb;
            if (d3 < d2 || (d3 == d2 && i3 < i2)) { db = d3; ib = i3; sb = 3; }
            else                                  { db = d2; ib = i2; sb = 2; }
            int win, widx;
            if (db < da || (db == da && ib < ia)) { win = sb; widx = ib; }
            else                                  { win = sa; widx = ia; }
            orow[j] = widx;
            if (win == 0) ++p0;
            else if (win == 1) ++p1;
            else if (win == 2) ++p2;
            else ++p3;
        }
    }
}

// ---------------------------------------------------------------------------
extern "C" void kernel_launch(void* const* d_in, const int* in_sizes, int n_in,
                              void* d_out, int out_size, void* d_ws, size_t ws_size,
                              hipStream_t stream) {
    (void)n_in; (void)out_size; (void)ws_size;
    const float* x = (const float*)d_in[0];
    const int N = in_sizes[0] / 64;         // 16000; N % 64 == 0

    float* ws          = (float*)d_ws;
    float* sq          = ws;                        // N
    float* dens        = ws + (size_t)N;            // N
    float* sqs         = ws + (size_t)2 * N;        // N (permuted norms)
    float* xs          = ws + (size_t)3 * N;        // N*64 (permuted points)
    unsigned* maxbits  = (unsigned*)(ws + (size_t)3 * N + (size_t)N * 64);

    float* dens_sorted = (float*)d_out;             // N floats
    int*   rips        = (int*)((float*)d_out + N); // N*15 int32 (bit-stored)

    hipMemsetAsync(maxbits, 0, sizeof(unsigned), stream);

    norms_kernel<<<(N + 255) / 256, 256, 0, stream>>>(x, sq, N);
    kde_kernel<<<N / 16, 128, 0, stream>>>(x, sq, dens, N);
    max_kernel<<<128, 256, 0, stream>>>(dens, maxbits, N);
    rank_kernel<<<(N + 255) / 256, 256, (size_t)N * sizeof(float), stream>>>(
        x, sq, dens, maxbits, xs, sqs, dens_sorted, N);
    rips_kernel<<<N / 16, 128, 0, stream>>>(xs, sqs, rips, N);
}